// BiLinearLayerV2_72378788872870
// MI455X (gfx1250) — compile-verified
//
#include <hip/hip_runtime.h>

typedef __attribute__((ext_vector_type(2))) float v2f;
typedef __attribute__((ext_vector_type(4))) float f4;
typedef __attribute__((ext_vector_type(8))) float v8f;

constexpr int F = 64;
constexpr int E = 32;

// ---------------------------------------------------------------------------
// Kernel 1: biLinear[b,f,d] = sum_e feature[b,f,e] * weight[f,e,d]
// One wave32 computes one 16x16 (batch x d) tile for one f, using fp32 WMMA
// V_WMMA_F32_16X16X4_F32 with a K=32 loop in steps of 4 (8 WMMAs, unrolled).
// ---------------------------------------------------------------------------
__global__ __launch_bounds__(128) void bilinear_wmma_f32(
    const float* __restrict__ feature,   // [B, F, E]
    const float* __restrict__ weight,    // [F, E, E]
    float* __restrict__ biLinear,        // [B, F, E] (workspace)
    int numBTiles)                       // B / 16
{
  const int lane  = threadIdx.x & 31;
  const int wid   = blockIdx.x * (blockDim.x >> 5) + (threadIdx.x >> 5);

  // wid -> (f, btile, ctile); E/16 == 2 column tiles per f
  const int tilesPerF = numBTiles * (E / 16);
  const int f     = wid / tilesPerF;
  const int rem   = wid % tilesPerF;
  const int b0    = (rem >> 1) * 16;   // batch-row tile base
  const int c0    = (rem & 1) * 16;    // d-column tile base

  const int half  = lane >> 4;         // 0 = lanes 0-15, 1 = lanes 16-31
  const int lid   = lane & 15;

  // A tile: row lid of feature[b0+lid, f, :]
  const float* Abase = feature + (size_t)(b0 + lid) * (F * E) + (size_t)f * E;
  // B tile: column (c0+lid) of weight[f, :, :]
  const float* Bbase = weight + (size_t)f * (E * E) + c0 + lid;

  v8f acc = {};
#pragma unroll
  for (int kb = 0; kb < E; kb += 4) {
    // A 16x4 layout: VGPR0 = A[r][kb+2*half], VGPR1 = A[r][kb+2*half+1]
    v2f a;
    a.x = Abase[kb + 2 * half + 0];
    a.y = Abase[kb + 2 * half + 1];
    // B 4x16 layout: VGPR0 = B[kb+2*half][c], VGPR1 = B[kb+2*half+1][c]
    v2f b;
    b.x = Bbase[(size_t)(kb + 2 * half + 0) * E];
    b.y = Bbase[(size_t)(kb + 2 * half + 1) * E];
    // 8 args: (neg_a, A, neg_b, B, c_mod, C, reuse_a, reuse_b)
    acc = __builtin_amdgcn_wmma_f32_16x16x4_f32(
        false, a, false, b, (short)0, acc, false, false);
  }

  // D 16x16 layout: VGPR i -> row (i + 8*half), col lid
  float* Dbase = biLinear + (size_t)f * E + c0 + lid;
#pragma unroll
  for (int i = 0; i < 8; ++i) {
    const int m = b0 + i + 8 * half;
    Dbase[(size_t)m * (F * E)] = acc[i];
  }
}

// ---------------------------------------------------------------------------
// Kernel 2: out[b,f,g,e] = biLinear[b,f,e] * feature[b,g,e] * weightLeft[f,g]
// Pure streaming store kernel: one float4 per thread, flat output index equals
// the thread's global vector index -> perfectly coalesced b128 stores.
// 268 MB of NT stores dominates (> 192 MB L2, so keep them out of cache);
// reads stay RT and hit L1/L2 with 64x reuse per cacheline.
// ---------------------------------------------------------------------------
__global__ __launch_bounds__(256) void outer_stream(
    const float* __restrict__ biLinear,   // [B, F, E]
    const float* __restrict__ feature,    // [B, F, E]
    const float* __restrict__ weightLeft, // [F, F]
    f4* __restrict__ out)                 // [B, F, F, E] viewed as float4
{
  const long idx = (long)blockIdx.x * blockDim.x + threadIdx.x; // float4 index
  // idx = (((b*F + f)*F + g)*E + e) / 4 ; E/4 = 8, F = 64
  const int  e4 = (int)(idx & 7);
  const int  g  = (int)((idx >> 3) & 63);
  const int  f  = (int)((idx >> 9) & 63);
  const long b  = idx >> 15;

  const f4 bl = *(const f4*)(biLinear + ((b * F + f) * E + e4 * 4));
  const f4 ft = *(const f4*)(feature  + ((b * F + g) * E + e4 * 4));
  const float wl = weightLeft[f * F + g];

  const f4 o = bl * ft * wl;
  // Non-temporal: output stream (268 MB) must not evict the read working set.
  __builtin_nontemporal_store(o, &out[idx]);
}

extern "C" void kernel_launch(void* const* d_in, const int* in_sizes, int n_in,
                              void* d_out, int out_size, void* d_ws, size_t ws_size,
                              hipStream_t stream) {
  (void)n_in; (void)out_size; (void)ws_size;

  const float* feature    = (const float*)d_in[0];  // [B, F, E]
  const float* weight     = (const float*)d_in[1];  // [F, E, E]
  const float* weightLeft = (const float*)d_in[2];  // [F, F]
  float* out      = (float*)d_out;                  // [B, F, F, E]
  float* biLinear = (float*)d_ws;                   // [B, F, E] scratch (4.2 MB)

  const int B = in_sizes[0] / (F * E);              // 512 for the reference
  const int numBTiles = B / 16;

  // Kernel 1: F * (B/16) * (E/16) wave-tiles, 4 waves (128 threads) per block.
  const int totalWaves = F * numBTiles * (E / 16);
  const int blocks1 = totalWaves / 4;
  bilinear_wmma_f32<<<blocks1, 128, 0, stream>>>(feature, weight, biLinear,
                                                 numBTiles);

  // Kernel 2: one float4 per thread over B*F*F*E elements.
  const long nvec4 = (long)B * F * F * E / 4;
  const long blocks2 = nvec4 / 256;
  outer_stream<<<dim3((unsigned)blocks2), 256, 0, stream>>>(
      biLinear, feature, weightLeft, (f4*)out);
}